// Net_85873576116919
// MI455X (gfx1250) — compile-verified
//
#include <hip/hip_runtime.h>

typedef float v2f __attribute__((ext_vector_type(2)));
typedef float v8f __attribute__((ext_vector_type(8)));

#define WAVES_PER_BLOCK 8
#define ROWS_PER_WAVE   32

// D = A(16x4 f32) * B(4x16 f32) + C(16x16 f32), full f32 precision.
static __device__ __forceinline__ v8f wmma_f32_16x16x4(v2f a, v2f b, v8f c) {
  return __builtin_amdgcn_wmma_f32_16x16x4_f32(
      /*neg_a=*/false, a, /*neg_b=*/false, b,
      /*c_mod=*/(short)0, c, /*reuse_a=*/false, /*reuse_b=*/false);
}

// Branchless tanh: prefer CDNA5 hardware V_TANH_F32.
static __device__ __forceinline__ float fast_tanh(float x) {
#if defined(__gfx1250__) && __has_builtin(__builtin_amdgcn_tanhf)
  return __builtin_amdgcn_tanhf(x);
#else
  // Branchless fallback: 1 - 2/(e^{2x}+1). __expf -> v_exp_f32.
  // Overflow -> inf -> returns 1; underflow -> 0 -> returns -1.
  float e = __expf(2.0f * x);
  return 1.0f - 2.0f / (e + 1.0f);
#endif
}

// Branchless padded gather of B[k][n] = W[n][k] (zero outside O x I):
// unconditional load from clamped in-bounds address, then cndmask-select 0.
static __device__ __forceinline__ float wElemT(const float* __restrict__ W,
                                               int O, int I, int k, int col) {
  int cc = col < O ? col : O - 1;
  int kk = k < I ? k : I - 1;
  float v = W[cc * I + kk];
  return (col < O && k < I) ? v : 0.0f;
}

static __device__ __forceinline__ float biasElem(const float* __restrict__ b,
                                                 int O, int col) {
  float v = b[col < O ? col : O - 1];
  return col < O ? v : 0.0f;
}

__global__ __launch_bounds__(256) void fk_mlp_wmma(
    const float* __restrict__ joints, const float* __restrict__ fk,
    const float* __restrict__ W1, const float* __restrict__ b1,
    const float* __restrict__ W2, const float* __restrict__ b2,
    const float* __restrict__ W3, const float* __restrict__ b3,
    float* __restrict__ out) {
  __shared__ float ldsJ[WAVES_PER_BLOCK][ROWS_PER_WAVE * 8]; // joints, padded to 8
  __shared__ float ldsH[WAVES_PER_BLOCK][2][16 * 16];        // activation tiles

  const int lane = threadIdx.x & 31;
  const int wave = threadIdx.x >> 5;
  const int rowBase = blockIdx.x * (WAVES_PER_BLOCK * ROWS_PER_WAVE) + wave * ROWS_PER_WAVE;

  // ---- Phase 0: stage 32 rows x 7 f32 of joints into LDS (coalesced) ----
  #pragma unroll
  for (int i = 0; i < 7; ++i) {
    int idx = i * 32 + lane; // 0..223
    float v = joints[rowBase * 7 + idx];
    ldsJ[wave][(idx / 7) * 8 + (idx % 7)] = v;
  }
  ldsJ[wave][lane * 8 + 7] = 0.0f; // K pad
  __syncthreads();

  const int nCol = lane & 15; // N index in B/C/D layouts
  const int half = lane >> 4;
  const int mRow = lane & 15; // M index in A layout (per 16-row tile)

  // ---- Weights in WMMA-B layout: k = 4*chunk + 2*half + v ----
  v2f B1m[2], B2m[4], B3m[4];
  #pragma unroll
  for (int c = 0; c < 2; ++c) {
    int k = 4 * c + 2 * half;
    B1m[c].x = wElemT(W1, 15, 7, k, nCol);
    B1m[c].y = wElemT(W1, 15, 7, k + 1, nCol);
  }
  #pragma unroll
  for (int c = 0; c < 4; ++c) {
    int k = 4 * c + 2 * half;
    B2m[c].x = wElemT(W2, 15, 15, k, nCol);
    B2m[c].y = wElemT(W2, 15, 15, k + 1, nCol);
    B3m[c].x = wElemT(W3, 7, 15, k, nCol);
    B3m[c].y = wElemT(W3, 7, 15, k + 1, nCol);
  }
  const float bias1 = biasElem(b1, 15, nCol);
  const float bias2 = biasElem(b2, 15, nCol);
  const float bias3 = biasElem(b3, 7, nCol);

  v8f Hv[2];

  // ---- Layer 1: joints(16x8) @ W1.T(8x16), 2 tiles x 2 K-chunks ----
  #pragma unroll
  for (int t = 0; t < 2; ++t) {
    v8f acc = {0.f, 0.f, 0.f, 0.f, 0.f, 0.f, 0.f, 0.f};
    #pragma unroll
    for (int c = 0; c < 2; ++c) {
      int k = 4 * c + 2 * half;
      v2f a;
      a.x = ldsJ[wave][(t * 16 + mRow) * 8 + k];
      a.y = ldsJ[wave][(t * 16 + mRow) * 8 + k + 1];
      acc = wmma_f32_16x16x4(a, B1m[c], acc);
    }
    #pragma unroll
    for (int r = 0; r < 8; ++r) acc[r] = fast_tanh(acc[r] + bias1);
    Hv[t] = acc;
  }
  __syncthreads();
  #pragma unroll
  for (int t = 0; t < 2; ++t)
    #pragma unroll
    for (int r = 0; r < 8; ++r)
      ldsH[wave][t][(r + 8 * half) * 16 + nCol] = Hv[t][r]; // D-layout -> row-major
  __syncthreads();

  // ---- Layer 2: h(16x16) @ W2.T(16x16), 2 tiles x 4 K-chunks ----
  #pragma unroll
  for (int t = 0; t < 2; ++t) {
    v8f acc = {0.f, 0.f, 0.f, 0.f, 0.f, 0.f, 0.f, 0.f};
    #pragma unroll
    for (int c = 0; c < 4; ++c) {
      int k = 4 * c + 2 * half;
      v2f a;
      a.x = ldsH[wave][t][mRow * 16 + k];
      a.y = ldsH[wave][t][mRow * 16 + k + 1];
      acc = wmma_f32_16x16x4(a, B2m[c], acc);
    }
    #pragma unroll
    for (int r = 0; r < 8; ++r) acc[r] = fast_tanh(acc[r] + bias2);
    Hv[t] = acc;
  }
  __syncthreads();
  #pragma unroll
  for (int t = 0; t < 2; ++t)
    #pragma unroll
    for (int r = 0; r < 8; ++r)
      ldsH[wave][t][(r + 8 * half) * 16 + nCol] = Hv[t][r];
  __syncthreads();

  // ---- Layer 3: h(16x16) @ W3.T(16x16) -> correction (cols 0..6 valid) ----
  #pragma unroll
  for (int t = 0; t < 2; ++t) {
    v8f acc = {0.f, 0.f, 0.f, 0.f, 0.f, 0.f, 0.f, 0.f};
    #pragma unroll
    for (int c = 0; c < 4; ++c) {
      int k = 4 * c + 2 * half;
      v2f a;
      a.x = ldsH[wave][t][mRow * 16 + k];
      a.y = ldsH[wave][t][mRow * 16 + k + 1];
      acc = wmma_f32_16x16x4(a, B3m[c], acc);
    }
    #pragma unroll
    for (int r = 0; r < 8; ++r) acc[r] = fast_tanh(acc[r] + bias3);
    Hv[t] = acc;
  }
  __syncthreads();
  #pragma unroll
  for (int t = 0; t < 2; ++t)
    #pragma unroll
    for (int r = 0; r < 8; ++r)
      ldsH[wave][t][(r + 8 * half) * 16 + nCol] = Hv[t][r];
  __syncthreads();

  // ---- FK chain: one lane = one batch element (all 32 lanes active) ----
  const int myRow = rowBase + lane;
  float theta[7];
  #pragma unroll
  for (int j = 0; j < 7; ++j)
    theta[j] = ldsJ[wave][lane * 8 + j] +
               ldsH[wave][lane >> 4][(lane & 15) * 16 + j];

  float e[3][4]; // rows 0..2 of ee; row 3 stays [0,0,0,1] (affine chain)
  #pragma unroll
  for (int j = 0; j < 7; ++j) {
    float al = fk[j * 4 + 0];
    float aP = fk[j * 4 + 1];
    float dP = fk[j * 4 + 2];
    float of = fk[j * 4 + 3];
    // |args| < ~8 here: native v_sin_f32/v_cos_f32 are accurate and branchless.
    float th = theta[j] + of;
    float st = __sinf(th), ct = __cosf(th);
    float sa = __sinf(al), ca = __cosf(al);
    float M[3][4] = {
        {ct, -st, 0.0f, aP},
        {st * ca, ct * ca, -sa, -dP * sa},
        {st * sa, ct * sa, ca, ca * dP}};
    if (j == 0) {
      #pragma unroll
      for (int r = 0; r < 3; ++r)
        #pragma unroll
        for (int c = 0; c < 4; ++c) e[r][c] = M[r][c];
    } else {
      #pragma unroll
      for (int r = 0; r < 3; ++r) {
        float e0 = e[r][0], e1 = e[r][1], e2 = e[r][2], e3 = e[r][3];
        e[r][0] = e0 * M[0][0] + e1 * M[1][0] + e2 * M[2][0];
        e[r][1] = e0 * M[0][1] + e1 * M[1][1] + e2 * M[2][1];
        e[r][2] = e0 * M[0][2] + e1 * M[1][2] + e2 * M[2][2];
        e[r][3] = e0 * M[0][3] + e1 * M[1][3] + e2 * M[2][3] + e3;
      }
    }
  }
  out[myRow * 3 + 0] = e[0][3];
  out[myRow * 3 + 1] = e[1][3];
  out[myRow * 3 + 2] = e[2][3];
}

extern "C" void kernel_launch(void* const* d_in, const int* in_sizes, int n_in,
                              void* d_out, int out_size, void* d_ws, size_t ws_size,
                              hipStream_t stream) {
  const float* joints = (const float*)d_in[0];
  const float* fk     = (const float*)d_in[1];
  const float* W1     = (const float*)d_in[2];
  const float* b1     = (const float*)d_in[3];
  const float* W2     = (const float*)d_in[4];
  const float* b2     = (const float*)d_in[5];
  const float* W3     = (const float*)d_in[6];
  const float* b3     = (const float*)d_in[7];
  float* out = (float*)d_out;

  const int B = in_sizes[0] / 7; // 262144
  const int rowsPerBlock = WAVES_PER_BLOCK * ROWS_PER_WAVE; // 256
  dim3 grid((B + rowsPerBlock - 1) / rowsPerBlock);
  fk_mlp_wmma<<<grid, dim3(256), 0, stream>>>(joints, fk, W1, b1, W2, b2, W3, b3, out);
}